// IntraRNN_36713380446501
// MI455X (gfx1250) — compile-verified
//
#include <hip/hip_runtime.h>
#include <hip/hip_bf16.h>
#include <math.h>

// Problem constants
#define B_ 1024
#define M_ 15
#define H_ 256
#define E_ 256
#define V_ 30000

// Output layout (flat, return order): output, hidden_new, embedded_input, gru_output, attn
#define OUT_HIDDEN 30720000LL
#define OUT_EMB    30982144LL
#define OUT_GRU    31244288LL
#define OUT_ATTN   31506432LL

typedef __attribute__((ext_vector_type(2))) float v2f;
typedef __attribute__((ext_vector_type(8))) float v8f;

__device__ __forceinline__ v8f wmma4(v2f a, v2f b, v8f c) {
  // V_WMMA_F32_16X16X4_F32 : D = A(16x4) * B(4x16) + C(16x16), f32 all the way
  return __builtin_amdgcn_wmma_f32_16x16x4_f32(
      /*neg_a=*/false, a, /*neg_b=*/false, b,
      /*c_mod=*/(short)0, c, /*reuse_a=*/false, /*reuse_b=*/false);
}

// A fragment (16x4) from LDS (row-major, leading dim ld), k-tile at k0.
// Lanes 0-15 hold rows 0-15 with K = k0..k0+1 ; lanes 16-31 hold K = k0+2..k0+3.
__device__ __forceinline__ v2f afrag_lds(const float* S, int ld, int k0) {
  const int lane = threadIdx.x & 31;
  const int row  = lane & 15;
  const int kk   = k0 + ((lane >> 4) << 1);
  v2f a; a.x = S[row * ld + kk]; a.y = S[row * ld + kk + 1];
  return a;
}

// B fragment (4x16) from a row-major (K x N) matrix, tile origin column n0.
__device__ __forceinline__ v2f bfrag_rm(const float* __restrict__ W, int ld, int n0, int k0) {
  const int lane = threadIdx.x & 31;
  const int col  = lane & 15;
  const int kk   = k0 + ((lane >> 4) << 1);
  v2f b;
  b.x = W[(long)kk * ld + n0 + col];
  b.y = W[(long)(kk + 1) * ld + n0 + col];
  return b;
}

// B fragment where logical B = W^T and W is (N x K) row-major (ldk = K).
__device__ __forceinline__ v2f bfrag_t(const float* __restrict__ W, int ldk, int n0, int k0) {
  const int lane = threadIdx.x & 31;
  const int col  = lane & 15;
  const int kk   = k0 + ((lane >> 4) << 1);
  v2f b;
  b.x = W[(long)(n0 + col) * ldk + kk];
  b.y = W[(long)(n0 + col) * ldk + kk + 1];
  return b;
}

// ---------------------------------------------------------------------------
// Kernel 1: per-batch attention. One block (8 waves) per batch.
// ---------------------------------------------------------------------------
__global__ void __launch_bounds__(256) attn_kernel(
    const float* __restrict__ hidden,        // (1,B,H) -> (B,H)
    const float* __restrict__ inter_output,  // (B,M,H)
    const int*   __restrict__ user_list,     // (B,)
    const float* __restrict__ inter_W,  const float* __restrict__ inter_b,
    const float* __restrict__ hidden_W, const float* __restrict__ hidden_b,
    const float* __restrict__ lt1_W,    const float* __restrict__ lt1_b,
    const float* __restrict__ scale_W,  const float* __restrict__ scale_b,
    float* __restrict__ ctx_ws,              // (B,H)
    float* __restrict__ attn_out)            // (B,M)
{
  __shared__ float S_io[16 * H_];        // inter_output tile, row 15 zero
  __shared__ float S_cat[16 * (2 * H_)]; // [hidden_t_a | inter_output_a]
  __shared__ float S_r[16 * H_];         // tanh result
  __shared__ float E_s[16];
  __shared__ float A_s[16];

  const int b = blockIdx.x;
  const int t = threadIdx.x;
  const int u = user_list[b];
  const float* Wi = inter_W  + (long)u * H_ * H_;
  const float* Wh = hidden_W + (long)u * H_ * H_;
  const float* L1 = lt1_W    + (long)u * 2 * H_ * H_;

  // Stage inter_output[b] (pad M=15 -> 16 with a zero row)
  for (int i = t; i < 16 * H_; i += 256) {
    const int r = i >> 8, c = i & 255;
    S_io[i] = (r < M_) ? inter_output[(long)b * M_ * H_ + r * H_ + c] : 0.0f;
  }
  // hidden_t_a (1 row): each thread owns one output column; broadcast to rows 0..14
  {
    float acc = hidden_b[u * H_ + t];
    const float* hv = hidden + (long)b * H_;
    for (int k = 0; k < H_; ++k) acc += hv[k] * Wh[k * H_ + t];
    for (int r = 0; r < M_; ++r) S_cat[r * (2 * H_) + t] = acc;
    S_cat[15 * (2 * H_) + t] = 0.0f;
  }
  __syncthreads();

  const int wave = t >> 5;
  const int lane = t & 31;
  const int row8 = (lane >> 4) << 3;
  const int col  = lane & 15;

  // GEMM1: inter_output_a = S_io(16x256) @ Wi(256x256) + inter_b -> S_cat[:,256:]
  for (int i = 0; i < 2; ++i) {
    const int n0 = (wave * 2 + i) * 16;
    v8f acc = {};
    for (int k0 = 0; k0 < H_; k0 += 4)
      acc = wmma4(afrag_lds(S_io, H_, k0), bfrag_rm(Wi, H_, n0, k0), acc);
    const float bias = inter_b[u * H_ + n0 + col];
    for (int r = 0; r < 8; ++r)
      S_cat[(r + row8) * (2 * H_) + H_ + n0 + col] = acc[r] + bias;
  }
  __syncthreads();

  // GEMM2: result = tanh(S_cat(16x512) @ L1(512x256) + lt1_b) -> S_r
  for (int i = 0; i < 2; ++i) {
    const int n0 = (wave * 2 + i) * 16;
    v8f acc = {};
    for (int k0 = 0; k0 < 2 * H_; k0 += 4)
      acc = wmma4(afrag_lds(S_cat, 2 * H_, k0), bfrag_rm(L1, H_, n0, k0), acc);
    const float bias = lt1_b[u * H_ + n0 + col];
    for (int r = 0; r < 8; ++r)
      S_r[(r + row8) * H_ + n0 + col] = tanhf(acc[r] + bias);
  }
  __syncthreads();

  // energies
  if (t < M_) {
    float e = scale_b[u];
    for (int h = 0; h < H_; ++h) e += S_r[t * H_ + h] * scale_W[u * H_ + h];
    E_s[t] = e;
  }
  __syncthreads();
  // softmax over M=15 (tiny, single lane)
  if (t == 0) {
    float mx = E_s[0];
    for (int m = 1; m < M_; ++m) mx = fmaxf(mx, E_s[m]);
    float s = 0.0f;
    for (int m = 0; m < M_; ++m) { const float e = expf(E_s[m] - mx); A_s[m] = e; s += e; }
    const float inv = 1.0f / s;
    for (int m = 0; m < M_; ++m) { A_s[m] *= inv; attn_out[(long)b * M_ + m] = A_s[m]; }
  }
  __syncthreads();
  // context = attn @ inter_output[b]
  {
    float c = 0.0f;
    for (int m = 0; m < M_; ++m) c += A_s[m] * S_io[m * H_ + t];
    ctx_ws[(long)b * H_ + t] = c;
  }
}

// ---------------------------------------------------------------------------
// Kernel 2: GRU gate GEMMs. One wave per 16x16 output tile of (B x 768).
// ---------------------------------------------------------------------------
__global__ void __launch_bounds__(32) gates_kernel(
    const float* __restrict__ emb,    // (B,E)
    const float* __restrict__ ctx,    // (B,H)
    const float* __restrict__ hidden, // (B,H)
    const float* __restrict__ Wih,    // (768,512)
    const float* __restrict__ Whh,    // (768,256)
    const float* __restrict__ bih, const float* __restrict__ bhh,
    float* __restrict__ ggi, float* __restrict__ ggh) // (B,768) each
{
  const int b0 = blockIdx.x * 16;
  const int n0 = blockIdx.y * 16;
  const int lane  = threadIdx.x & 31;
  const int row   = lane & 15;
  const int kslot = (lane >> 4) << 1;
  const int col   = lane & 15;
  const int row8  = (lane >> 4) << 3;
  const int b = b0 + row;

  // gi = x @ Wih^T, x = [emb | ctx] (K=512)
  v8f gi = {};
  for (int k0 = 0; k0 < 2 * E_; k0 += 4) {
    const int kk = k0 + kslot; // pairs never straddle 256 (k0 % 4 == 0)
    const float* xs = (kk < E_) ? (emb + (long)b * E_ + kk)
                                : (ctx + (long)b * H_ + (kk - E_));
    v2f a; a.x = xs[0]; a.y = xs[1];
    gi = wmma4(a, bfrag_t(Wih, 2 * E_, n0, k0), gi);
  }
  // gh = h @ Whh^T (K=256)
  v8f gh = {};
  for (int k0 = 0; k0 < H_; k0 += 4) {
    const int kk = k0 + kslot;
    v2f a; a.x = hidden[(long)b * H_ + kk]; a.y = hidden[(long)b * H_ + kk + 1];
    gh = wmma4(a, bfrag_t(Whh, H_, n0, k0), gh);
  }
  const float bi = bih[n0 + col];
  const float bh = bhh[n0 + col];
  for (int r = 0; r < 8; ++r) {
    const long bb = b0 + r + row8;
    ggi[bb * 768 + n0 + col] = gi[r] + bi;
    ggh[bb * 768 + n0 + col] = gh[r] + bh;
  }
}

// ---------------------------------------------------------------------------
// Kernel 3: GRU gate fusion + aux outputs.
// ---------------------------------------------------------------------------
__global__ void __launch_bounds__(256) hnew_kernel(
    const float* __restrict__ ggi, const float* __restrict__ ggh,
    const float* __restrict__ hidden, const float* __restrict__ emb,
    float* __restrict__ hnew_ws, float* __restrict__ out)
{
  const int i = blockIdx.x * 256 + threadIdx.x; // 0 .. B*H-1
  const int b = i >> 8, j = i & 255;
  const long base = (long)b * 768 + j;
  const float r = 1.0f / (1.0f + expf(-(ggi[base]       + ggh[base])));
  const float z = 1.0f / (1.0f + expf(-(ggi[base + 256] + ggh[base + 256])));
  const float n = tanhf(ggi[base + 512] + r * ggh[base + 512]);
  const float h = hidden[i];
  const float hn = (1.0f - z) * n + z * h;
  hnew_ws[i] = hn;
  out[OUT_HIDDEN + i] = hn;     // hidden_new (1,B,H)
  out[OUT_GRU + i]    = hn;     // gru_output (B,1,H)
  out[OUT_EMB + i]    = emb[i]; // embedded_input passthrough
}

// ---------------------------------------------------------------------------
// Kernel 4: output = h_new (B x 256) @ lin_W (256 x 30000) + lin_b.
// 4 waves/block, each owns one 16-wide N tile; h_new tile staged in LDS.
// ---------------------------------------------------------------------------
__global__ void __launch_bounds__(128) final_kernel(
    const float* __restrict__ hnew,  // (B,H)
    const float* __restrict__ lin_W, // (H,V)
    const float* __restrict__ lin_b, // (V,)
    float* __restrict__ out)
{
  __shared__ float S_h[16 * H_];
  const int b0 = blockIdx.x * 16;
  const int t = threadIdx.x;
  for (int i = t; i < 16 * H_; i += 128)
    S_h[i] = hnew[(long)b0 * H_ + i];
  __syncthreads();

  const int wave = t >> 5;
  const int nt = blockIdx.y * 4 + wave;       // 30000/16 = 1875 N tiles
  if (nt >= V_ / 16) return;                  // whole-wave exit: EXEC stays full
  const int n0 = nt * 16;
  const int lane = t & 31;
  const int col  = lane & 15;
  const int row8 = (lane >> 4) << 3;

  v8f acc = {};
  for (int k0 = 0; k0 < H_; k0 += 4)
    acc = wmma4(afrag_lds(S_h, H_, k0), bfrag_rm(lin_W, V_, n0, k0), acc);

  const float bias = lin_b[n0 + col];
  for (int r = 0; r < 8; ++r)
    out[(long)(b0 + r + row8) * V_ + n0 + col] = acc[r] + bias;
}

// ---------------------------------------------------------------------------
extern "C" void kernel_launch(void* const* d_in, const int* in_sizes, int n_in,
                              void* d_out, int out_size, void* d_ws, size_t ws_size,
                              hipStream_t stream) {
  const float* emb    = (const float*)d_in[1];   // input_embedding (B,1,E)
  const float* hidden = (const float*)d_in[2];   // hidden (1,B,H)
  const float* io     = (const float*)d_in[3];   // inter_output (B,M,H)
  const int*   users  = (const int*)  d_in[4];   // user_list (B,)
  const float* iW  = (const float*)d_in[5];
  const float* ib  = (const float*)d_in[6];
  const float* hW  = (const float*)d_in[7];
  const float* hb  = (const float*)d_in[8];
  const float* l1W = (const float*)d_in[9];
  const float* l1b = (const float*)d_in[10];
  const float* sW  = (const float*)d_in[11];
  const float* sb  = (const float*)d_in[12];
  const float* gWih = (const float*)d_in[13];
  const float* gWhh = (const float*)d_in[14];
  const float* gbih = (const float*)d_in[15];
  const float* gbhh = (const float*)d_in[16];
  const float* linW = (const float*)d_in[17];
  const float* linb = (const float*)d_in[18];

  float* out = (float*)d_out;
  float* ws  = (float*)d_ws;
  float* ctx  = ws;                  // B*H      = 262144 floats
  float* ggi  = ctx + 262144;        // B*768    = 786432
  float* ggh  = ggi + 786432;        // B*768
  float* hnew = ggh + 786432;        // B*H      -> total 8 MB

  attn_kernel<<<B_, 256, 0, stream>>>(hidden, io, users, iW, ib, hW, hb,
                                      l1W, l1b, sW, sb, ctx, out + OUT_ATTN);
  gates_kernel<<<dim3(B_ / 16, 768 / 16), 32, 0, stream>>>(
      emb, ctx, hidden, gWih, gWhh, gbih, gbhh, ggi, ggh);
  hnew_kernel<<<(B_ * H_) / 256, 256, 0, stream>>>(ggi, ggh, hidden, emb, hnew, out);
  final_kernel<<<dim3(B_ / 16, (V_ / 16 + 3) / 4), 128, 0, stream>>>(hnew, linW, linb, out);
}